// NormalEstimator_35880156790881
// MI455X (gfx1250) — compile-verified
//
#include <hip/hip_runtime.h>
#include <math.h>

typedef __attribute__((ext_vector_type(2))) float v2f;
typedef __attribute__((ext_vector_type(8))) float v8f;

#define NPTS      8192
#define KNN       30
#define NTILE     16
#define NITER     (NPTS / NTILE)       // 512
#define ROWSTRIDE 20                   // padded dwords per scratch row (bank-conflict free)
#define PTS_FLOATS (NPTS * 4)          // 32768 floats = 128 KB (x,y,z,|p|^2)
#define SCR_FLOATS (32 * ROWSTRIDE)    // 640 floats per wave

// One block = 256 threads = 8 wave32, owns 256 query points of one batch.
// Each wave owns 32 query rows (two 16-row WMMA M-tiles).
extern "C" __global__ __launch_bounds__(256)
void normal_est_kernel(const float* __restrict__ x, float* __restrict__ out)
{
    extern __shared__ float smem[];
    float* pts = smem;                         // float4 per point
    const int tid   = threadIdx.x;
    const int wave  = tid >> 5;
    const int lane  = tid & 31;
    const int b     = blockIdx.x >> 5;         // 32 blocks per batch
    const int q0blk = (blockIdx.x & 31) * 256;
    const float* __restrict__ xb = x + b * (3 * NPTS);

    // ---- stage all points of this batch into LDS: (x, y, z, |p|^2) ----
    for (int j = tid; j < NPTS; j += 256) {
        float px = xb[j];
        float py = xb[NPTS + j];
        float pz = xb[2 * NPTS + j];
        float n2 = px * px + py * py + pz * pz;
        float4 v; v.x = px; v.y = py; v.z = pz; v.w = n2;
        ((float4*)pts)[j] = v;
    }
    __syncthreads();

    const float4* pts4 = (const float4*)pts;
    float* scr = smem + PTS_FLOATS + wave * SCR_FLOATS;   // per-wave scratch

    const int  qwave = q0blk + wave * 32;   // first of this wave's 32 rows
    const bool hi    = (lane >= 16);
    const int  ln    = lane & 15;

    // Loop-invariant A fragments for the two 16-row tiles.
    // A(16x4) f32 layout: lanes 0-15 hold K=0,1 ; lanes 16-31 hold K=2,3 (same M = ln).
    // A row m = (xm, ym, zm, 1)
    float4 pa0 = pts4[qwave + ln];
    float4 pa1 = pts4[qwave + 16 + ln];
    v2f a0, a1;
    a0.x = hi ? pa0.z : pa0.x;   a0.y = hi ? 1.0f : pa0.y;
    a1.x = hi ? pa1.z : pa1.x;   a1.y = hi ? 1.0f : pa1.y;

    // top-30 candidate lists, sorted ascending, register resident
    float bd[KNN]; int bi[KNN];
#pragma unroll
    for (int s = 0; s < KNN; ++s) { bd[s] = 3.4e38f; bi[s] = 0; }

    for (int jt = 0; jt < NITER; ++jt) {
        const int jbase = jt * NTILE;
        // B(4x16) col j = (-2xj, -2yj, -2zj, |pj|^2); lanes 0-15: K=0,1 ; lanes 16-31: K=2,3
        float4 pb = pts4[jbase + ln];
        v2f bv;
        bv.x = hi ? (-2.0f * pb.z) : (-2.0f * pb.x);
        bv.y = hi ?  pb.w          : (-2.0f * pb.y);

        v8f c0 = {0.f, 0.f, 0.f, 0.f, 0.f, 0.f, 0.f, 0.f};
        // D(i,j) = |pj|^2 - 2 pi.pj   (row-constant |pi|^2 dropped: irrelevant for top-k)
        v8f d0 = __builtin_amdgcn_wmma_f32_16x16x4_f32(false, a0, false, bv, (short)0, c0, false, false);
        v8f d1 = __builtin_amdgcn_wmma_f32_16x16x4_f32(false, a1, false, bv, (short)0, c0, false, false);

        // spill D tiles row-major (padded) so each lane can scan its own row
        const int mrow = hi ? 8 : 0;
#pragma unroll
        for (int r = 0; r < 8; ++r) {
            scr[(mrow + r) * ROWSTRIDE + ln]      = d0[r];
            scr[(16 + mrow + r) * ROWSTRIDE + ln] = d1[r];
        }

        // each lane scans its row's 16 candidates (LDS ops are in-order per wave)
        const float4* rowv = (const float4*)(scr + lane * ROWSTRIDE);
#pragma unroll
        for (int g = 0; g < 4; ++g) {
            float4 dv = rowv[g];
            float dd[4] = { dv.x, dv.y, dv.z, dv.w };
#pragma unroll
            for (int e = 0; e < 4; ++e) {
                float d = dd[e];
                int   j = jbase + g * 4 + e;
                if (d < bd[KNN - 1]) {          // threshold reject (common case)
                    float cd = d; int ci = j;   // branchless sorted insert, static indices
#pragma unroll
                    for (int s = 0; s < KNN; ++s) {
                        bool lt = cd < bd[s];
                        float td = bd[s]; int ti = bi[s];
                        bd[s] = lt ? cd : td;  bi[s] = lt ? ci : ti;
                        cd    = lt ? td : cd;  ci    = lt ? ti : ci;
                    }
                }
            }
        }
    }

    // ---- covariance of 30 neighbors ----
    float sx = 0, sy = 0, sz = 0;
    float Sxx = 0, Sxy = 0, Sxz = 0, Syy = 0, Syz = 0, Szz = 0;
#pragma unroll
    for (int s = 0; s < KNN; ++s) {
        float4 pp = pts4[bi[s]];
        sx += pp.x; sy += pp.y; sz += pp.z;
        Sxx += pp.x * pp.x; Sxy += pp.x * pp.y; Sxz += pp.x * pp.z;
        Syy += pp.y * pp.y; Syz += pp.y * pp.z; Szz += pp.z * pp.z;
    }
    const float kk = (float)KNN;
    float mx = sx / kk, my = sy / kk, mz = sz / kk;
    float a00 = Sxx - kk * mx * mx, a01 = Sxy - kk * mx * my, a02 = Sxz - kk * mx * mz;
    float a11 = Syy - kk * my * my, a12 = Syz - kk * my * mz, a22 = Szz - kk * mz * mz;

    // ---- smallest eigenvalue of symmetric 3x3 (trigonometric method) ----
    float tr  = a00 + a11 + a22;
    float qq  = tr * (1.0f / 3.0f);
    float b00 = a00 - qq, b11 = a11 - qq, b22 = a22 - qq;
    float p1  = a01 * a01 + a02 * a02 + a12 * a12;
    float p2  = b00 * b00 + b11 * b11 + b22 * b22 + 2.0f * p1;
    float p   = sqrtf(p2 * (1.0f / 6.0f));
    float invp = 1.0f / fmaxf(p, 1e-12f);
    float c00 = b00 * invp, c11 = b11 * invp, c22 = b22 * invp;
    float c01 = a01 * invp, c02 = a02 * invp, c12 = a12 * invp;
    float hdet = 0.5f * (c00 * (c11 * c22 - c12 * c12)
                       - c01 * (c01 * c22 - c12 * c02)
                       + c02 * (c01 * c12 - c11 * c02));
    hdet = fminf(fmaxf(hdet, -1.0f), 1.0f);
    float phi  = acosf(hdet) * (1.0f / 3.0f);
    float lmin = qq + 2.0f * p * cosf(phi + 2.09439510239319549f); // +2*pi/3

    // ---- eigenvector of lmin: best cross product of rows of (A - lmin I) ----
    float r0x = a00 - lmin, r0y = a01,        r0z = a02;
    float r1x = a01,        r1y = a11 - lmin, r1z = a12;
    float r2x = a02,        r2y = a12,        r2z = a22 - lmin;
    float v0x = r0y * r1z - r0z * r1y, v0y = r0z * r1x - r0x * r1z, v0z = r0x * r1y - r0y * r1x;
    float v1x = r0y * r2z - r0z * r2y, v1y = r0z * r2x - r0x * r2z, v1z = r0x * r2y - r0y * r2x;
    float v2x = r1y * r2z - r1z * r2y, v2y = r1z * r2x - r1x * r2z, v2z = r1x * r2y - r1y * r2x;
    float n0 = v0x * v0x + v0y * v0y + v0z * v0z;
    float n1 = v1x * v1x + v1y * v1y + v1z * v1z;
    float n2 = v2x * v2x + v2y * v2y + v2z * v2z;
    float bx = v0x, by = v0y, bz = v0z, bn = n0;
    if (n1 > bn) { bx = v1x; by = v1y; bz = v1z; bn = n1; }
    if (n2 > bn) { bx = v2x; by = v2y; bz = v2z; bn = n2; }
    if (bn < 1e-30f) { bx = 0.0f; by = 0.0f; bz = 1.0f; bn = 1.0f; }
    float inv = rsqrtf(bn);
    float nx = bx * inv, ny = by * inv, nz = bz * inv;

    // ---- orient toward origin: flip if dot(-p, n) < 0 ----
    float4 qp = hi ? pa1 : pa0;   // this lane's own query point
    float dotv = -(qp.x * nx + qp.y * ny + qp.z * nz);
    if (dotv < 0.0f) { nx = -nx; ny = -ny; nz = -nz; }

    // ---- write output (4,6,8192): rows 0-2 input copy, rows 3-5 normals ----
    float* ob = out + b * (6 * NPTS);
    const int q = qwave + lane;   // == q0blk + tid
    ob[0 * NPTS + q] = qp.x;
    ob[1 * NPTS + q] = qp.y;
    ob[2 * NPTS + q] = qp.z;
    ob[3 * NPTS + q] = nx;
    ob[4 * NPTS + q] = ny;
    ob[5 * NPTS + q] = nz;
}

extern "C" void kernel_launch(void* const* d_in, const int* in_sizes, int n_in,
                              void* d_out, int out_size, void* d_ws, size_t ws_size,
                              hipStream_t stream) {
    (void)in_sizes; (void)n_in; (void)out_size; (void)d_ws; (void)ws_size;
    const float* x = (const float*)d_in[0];   // (4,3,8192) f32 ; d_in[1] is k==30 (hardcoded)
    float* out = (float*)d_out;               // (4,6,8192) f32
    const size_t shmem = (size_t)(PTS_FLOATS + 8 * SCR_FLOATS) * sizeof(float); // 151552 B
    normal_est_kernel<<<dim3(4 * 32), dim3(256), shmem, stream>>>(x, out);
}